// GCNLayer_34522947125341
// MI455X (gfx1250) — compile-verified
//
#include <hip/hip_runtime.h>
#include <hip/hip_bf16.h>

typedef __attribute__((ext_vector_type(2))) float v2f;
typedef __attribute__((ext_vector_type(8))) float v8f;

#define F_IN  128
#define F_OUT 64

// ---------------- degree ----------------
__global__ void k_init_deg(float* deg, int N) {
    int i = blockIdx.x * blockDim.x + threadIdx.x;
    if (i < N) deg[i] = 1.0f;  // self-loop counts once
}

__global__ void k_edge_deg(const long long* __restrict__ src, float* deg, int E) {
    int e = blockIdx.x * blockDim.x + threadIdx.x;
    if (e < E) atomicAdd(&deg[(int)src[e]], 1.0f);
}

__global__ void k_dinv(float* deg, int N) {
    int i = blockIdx.x * blockDim.x + threadIdx.x;
    if (i < N) deg[i] = rsqrtf(deg[i]);  // deg >= 1 always
}

// ---------------- GEMM h = x @ W^T via V_WMMA_F32_16X16X4_F32 ----------------
// block = 128 threads = 4 waves; wave w computes the 16x16 tile at
// rows m0 = 16*blockIdx.x, cols n0 = 16*w. K accumulated in steps of 4.
__global__ void k_gemm_wmma(const float* __restrict__ x, const float* __restrict__ W,
                            float* __restrict__ h) {
    const int wave = threadIdx.x >> 5;
    const int lane = threadIdx.x & 31;
    const int half = lane >> 4;       // 0: K,K+1 ; 1: K+2,K+3
    const int l15  = lane & 15;
    const int m0 = blockIdx.x * 16;
    const int n0 = wave * 16;

    const float* xrow = x + (size_t)(m0 + l15) * F_IN;   // A row (M = l15)
    const float* wrow = W + (size_t)(n0 + l15) * F_IN;   // B col as W row (N = l15)

    v8f c = {};
    #pragma unroll 4
    for (int k = 0; k < F_IN; k += 4) {
        const int ka = k + 2 * half;
        v2f a = *(const v2f*)(xrow + ka);   // A[M=l15][ka], [ka+1]
        v2f b = *(const v2f*)(wrow + ka);   // B[K=ka][N=l15] = W[n][ka], [ka+1]
        c = __builtin_amdgcn_wmma_f32_16x16x4_f32(
                /*neg_a=*/false, a, /*neg_b=*/false, b,
                /*c_mod=*/(short)0, c, /*reuse_a=*/false, /*reuse_b=*/false);
    }

    // C/D layout: VGPR r -> M = r (lanes 0-15) / r+8 (lanes 16-31), N = lane&15
    float* outp = h + (size_t)(m0 + half * 8) * F_OUT + n0 + l15;
    #pragma unroll
    for (int r = 0; r < 8; ++r)
        outp[(size_t)r * F_OUT] = c[r];
}

// ---------------- aggregation ----------------
__global__ void k_zero_stats(float* stats) {  // sums[64] + sumsq[64]
    if (threadIdx.x < 128) stats[threadIdx.x] = 0.0f;
}

// out[i,:] = h[i,:] * dinv[i]^2   (self-loop term; runs before edge scatter)
__global__ void k_selfloop(const float* __restrict__ h, const float* __restrict__ dinv,
                           float* __restrict__ out, long long total) {
    long long g = (long long)blockIdx.x * blockDim.x + threadIdx.x;
    if (g < total) {
        int i = (int)(g >> 6);
        float di = dinv[i];
        out[g] = h[g] * di * di;
    }
}

// 32 lanes per edge, 2 features each: out[dst,:] += h[src,:] * norm
__global__ void k_edge_agg(const long long* __restrict__ ei, const float* __restrict__ ew,
                           const float* __restrict__ h, const float* __restrict__ dinv,
                           float* __restrict__ out, int E) {
    long long g = (long long)blockIdx.x * blockDim.x + threadIdx.x;
    int e    = (int)(g >> 5);
    int lane = (int)(g & 31);
    if (e >= E) return;
    int s = (int)ei[e];
    int d = (int)ei[e + E];
    float norm = dinv[s] * ew[e] * dinv[d];
    float2 v = ((const float2*)(h + (size_t)s * F_OUT))[lane];
    float* o = out + (size_t)d * F_OUT + lane * 2;
    atomicAdd(o,     v.x * norm);
    atomicAdd(o + 1, v.y * norm);
}

// ---------------- BN stats over relu(out) ----------------
__global__ void k_stats(const float* __restrict__ out, float* sums, float* sumsq, int N) {
    __shared__ float ssum[256], ssq[256];
    const int col = threadIdx.x & 63;
    const int rg  = threadIdx.x >> 6;  // 0..3 row groups
    float s = 0.0f, q = 0.0f;
    for (int row = blockIdx.x * 4 + rg; row < N; row += gridDim.x * 4) {
        float v = out[(size_t)row * F_OUT + col];
        v = v > 0.0f ? v : 0.0f;
        s += v; q += v * v;
    }
    ssum[threadIdx.x] = s; ssq[threadIdx.x] = q;
    __syncthreads();
    if (threadIdx.x < 64) {
        s = ssum[threadIdx.x] + ssum[threadIdx.x + 64] + ssum[threadIdx.x + 128] + ssum[threadIdx.x + 192];
        q = ssq [threadIdx.x] + ssq [threadIdx.x + 64] + ssq [threadIdx.x + 128] + ssq [threadIdx.x + 192];
        atomicAdd(&sums[threadIdx.x],  s);
        atomicAdd(&sumsq[threadIdx.x], q);
    }
}

__global__ void k_finalize(const float* sums, const float* sumsq,
                           const float* gamma, const float* beta,
                           float* scale, float* shift, int N) {
    int c = threadIdx.x;  // 64 threads
    float invN = 1.0f / (float)N;
    float mean = sums[c] * invN;
    float var  = sumsq[c] * invN - mean * mean;
    float inv  = rsqrtf(var + 1e-5f);
    float sc   = inv * gamma[c];
    scale[c] = sc;
    shift[c] = beta[c] - mean * sc;
}

__global__ void k_apply(float* __restrict__ out, const float* __restrict__ scale,
                        const float* __restrict__ shift, long long total) {
    long long g = (long long)blockIdx.x * blockDim.x + threadIdx.x;
    if (g < total) {
        int c = (int)(g & 63);
        float v = out[g];
        v = v > 0.0f ? v : 0.0f;
        out[g] = v * scale[c] + shift[c];
    }
}

extern "C" void kernel_launch(void* const* d_in, const int* in_sizes, int n_in,
                              void* d_out, int out_size, void* d_ws, size_t ws_size,
                              hipStream_t stream) {
    const float*     x     = (const float*)d_in[0];      // [N, 128]
    const long long* ei    = (const long long*)d_in[1];  // [2, E] int64
    const float*     ew    = (const float*)d_in[2];      // [E]
    const float*     W     = (const float*)d_in[3];      // [64, 128]
    const float*     gamma = (const float*)d_in[4];      // [64]
    const float*     beta  = (const float*)d_in[5];      // [64]
    float*           out   = (float*)d_out;              // [N, 64]

    const int N = in_sizes[0] / F_IN;
    const int E = in_sizes[2];

    // workspace layout (floats)
    float* ws    = (float*)d_ws;
    float* h     = ws;                          // N*64
    float* deg   = ws + (size_t)N * F_OUT;      // N  (becomes dinv in place)
    float* sums  = deg + N;                     // 64
    float* sumsq = sums + 64;                   // 64
    float* scale = sums + 128;                  // 64
    float* shift = sums + 192;                  // 64

    const long long total = (long long)N * F_OUT;

    // 1-3: degree -> dinv
    k_init_deg<<<(N + 255) / 256, 256, 0, stream>>>(deg, N);
    k_edge_deg<<<(E + 255) / 256, 256, 0, stream>>>(ei, deg, E);
    k_dinv<<<(N + 255) / 256, 256, 0, stream>>>(deg, N);

    // 4: h = x @ W^T (WMMA f32). N is a multiple of 16 (100000/16 = 6250).
    k_gemm_wmma<<<N / 16, 128, 0, stream>>>(x, W, h);

    // 5-7: aggregation
    k_zero_stats<<<1, 128, 0, stream>>>(sums);
    k_selfloop<<<(int)((total + 255) / 256), 256, 0, stream>>>(h, deg, out, total);
    {
        long long work = (long long)E * 32;
        k_edge_agg<<<(int)((work + 255) / 256), 256, 0, stream>>>(ei, ew, h, deg, out, E);
    }

    // 8-10: batchnorm (training-mode batch stats) fused with ReLU
    k_stats<<<1024, 256, 0, stream>>>(out, sums, sumsq, N);
    k_finalize<<<1, 64, 0, stream>>>(sums, sumsq, gamma, beta, scale, shift, N);
    k_apply<<<(int)((total + 255) / 256), 256, 0, stream>>>(out, scale, shift, total);
}